// TrafficGCN_59150289600863
// MI455X (gfx1250) — compile-verified
//
#include <hip/hip_runtime.h>
#include <hip/hip_bf16.h>

#define F_IN  256
#define F_HID 16
#define F_OUT 3
#define BN_EPS 1e-5f

typedef __attribute__((ext_vector_type(2))) float v2f;
typedef __attribute__((ext_vector_type(8))) float v8f;

__device__ __forceinline__ void gatomic_add(float* p, float v) {
    __hip_atomic_fetch_add(p, v, __ATOMIC_RELAXED, __HIP_MEMORY_SCOPE_AGENT);
}

// ---------------- degree / normalization ----------------
__global__ void k_init_deg(float* __restrict__ deg, int n) {
    int i = blockIdx.x * blockDim.x + threadIdx.x;
    if (i < n) deg[i] = 1.0f;                     // self-loop
}

__global__ void k_accum_deg(const long long* __restrict__ dst, float* __restrict__ deg, int e) {
    int i = blockIdx.x * blockDim.x + threadIdx.x;
    if (i < e) gatomic_add(&deg[dst[i]], 1.0f);
}

__global__ void k_dinv(float* __restrict__ deg, int n) {
    int i = blockIdx.x * blockDim.x + threadIdx.x;
    if (i < n) deg[i] = rsqrtf(deg[i]);           // in place: deg -> dinv
}

// ---------------- GEMM1: m1[N x 16] = x[N x 256] @ W1[256 x 16] (WMMA f32 16x16x4) ----
__global__ void k_gemm1_wmma(const float* __restrict__ x, const float* __restrict__ w1,
                             float* __restrict__ m1, int n, int nTiles) {
    int gwave = (int)((blockIdx.x * (unsigned)blockDim.x + threadIdx.x) >> 5);
    if (gwave >= nTiles) return;                  // uniform per wave
    int lane = threadIdx.x & 31;
    int mrow = lane & 15;                         // M (for A) / N column (for B)
    int kh   = (lane >> 4) * 2;                   // K sub-offset: 0 or 2

    int arow = gwave * 16 + mrow;
    if (arow >= n) arow = n - 1;                  // clamp loads (full tiles when n%16==0)
    const float* xr = x  + (size_t)arow * F_IN + kh;   // A: x[arow][k+kh], x[arow][k+kh+1]
    const float* wc = w1 + (size_t)kh * F_HID + mrow;  // B: w1[k+kh][col], w1[k+kh+1][col]

    v8f acc = {};
#pragma unroll 8
    for (int k = 0; k < F_IN; k += 4) {
        v2f a, b;
        a.x = xr[k];
        a.y = xr[k + 1];
        b.x = wc[k * F_HID];
        b.y = wc[k * F_HID + F_HID];
        acc = __builtin_amdgcn_wmma_f32_16x16x4_f32(
            /*neg_a=*/false, a, /*neg_b=*/false, b,
            /*c_mod=*/(short)0, acc, /*reuse_a=*/false, /*reuse_b=*/false);
    }

    int rbase = gwave * 16 + (lane >> 4) * 8;     // C/D layout: VGPR r -> M = r + 8*(lane>=16)
#pragma unroll
    for (int r = 0; r < 8; ++r) {
        int row = rbase + r;
        if (row < n) m1[(size_t)row * F_HID + mrow] = acc[r];
    }
}

// ---------------- scatter conv1 (16 features) ----------------
__global__ void k_selfloop16(const float* __restrict__ m1, const float* __restrict__ dinv,
                             float* __restrict__ agg, int n) {
    int t = blockIdx.x * blockDim.x + threadIdx.x;
    int i = t >> 2, q = (t & 3) * 4;
    if (i < n) {
        float w = dinv[i] * dinv[i];
        float4 v = *(const float4*)(m1 + (size_t)i * F_HID + q);
        float4 o = { v.x * w, v.y * w, v.z * w, v.w * w };
        *(float4*)(agg + (size_t)i * F_HID + q) = o;
    }
}

__global__ void k_edge16(const long long* __restrict__ src, const long long* __restrict__ dst,
                         const float* __restrict__ m1, const float* __restrict__ dinv,
                         float* __restrict__ agg, int e) {
    int t = blockIdx.x * blockDim.x + threadIdx.x;
    int idx = t >> 2, q = (t & 3) * 4;
    if (idx < e) {
        long long s = src[idx], d = dst[idx];
        float w = dinv[s] * dinv[d];
        float4 v = *(const float4*)(m1 + (size_t)s * F_HID + q);
        float* ap = agg + (size_t)d * F_HID + q;
        gatomic_add(ap + 0, v.x * w);
        gatomic_add(ap + 1, v.y * w);
        gatomic_add(ap + 2, v.z * w);
        gatomic_add(ap + 3, v.w * w);
    }
}

// ---------------- batchnorm ----------------
__global__ void k_zero32(float* __restrict__ sums) {
    if (threadIdx.x < 32) sums[threadIdx.x] = 0.0f;
}

__global__ void k_bn_stats(const float* __restrict__ h, float* __restrict__ sums, int n) {
    __shared__ float ss[32];
    if (threadIdx.x < 32) ss[threadIdx.x] = 0.0f;
    __syncthreads();
    float ls[F_HID], lq[F_HID];
#pragma unroll
    for (int f = 0; f < F_HID; ++f) { ls[f] = 0.0f; lq[f] = 0.0f; }
    for (int i = blockIdx.x * blockDim.x + threadIdx.x; i < n; i += gridDim.x * blockDim.x) {
        const float4* r = (const float4*)(h + (size_t)i * F_HID);
#pragma unroll
        for (int q = 0; q < 4; ++q) {
            float4 v = r[q];
            ls[4*q+0] += v.x; lq[4*q+0] += v.x * v.x;
            ls[4*q+1] += v.y; lq[4*q+1] += v.y * v.y;
            ls[4*q+2] += v.z; lq[4*q+2] += v.z * v.z;
            ls[4*q+3] += v.w; lq[4*q+3] += v.w * v.w;
        }
    }
#pragma unroll
    for (int f = 0; f < F_HID; ++f) {
        atomicAdd(&ss[f], ls[f]);                 // ds_add_f32
        atomicAdd(&ss[F_HID + f], lq[f]);
    }
    __syncthreads();
    if (threadIdx.x < 32) gatomic_add(&sums[threadIdx.x], ss[threadIdx.x]);
}

__global__ void k_bn_relu(float* __restrict__ h, const float* __restrict__ sums,
                          const float* __restrict__ gamma, const float* __restrict__ beta,
                          int n, float invN) {
    int t = blockIdx.x * blockDim.x + threadIdx.x;
    int i = t >> 2, q = (t & 3) * 4;
    if (i < n) {
        float* hp = h + (size_t)i * F_HID + q;
#pragma unroll
        for (int j = 0; j < 4; ++j) {
            int f = q + j;
            float mu  = sums[f] * invN;
            float var = sums[F_HID + f] * invN - mu * mu;
            float rs  = rsqrtf(var + BN_EPS);
            float v   = (hp[j] - mu) * rs * gamma[f] + beta[f];
            hp[j] = fmaxf(v, 0.0f);
        }
    }
}

// ---------------- GEMM2: m2[N x 3] = h[N x 16] @ W2[16 x 3] ----------------
__global__ void k_gemm2(const float* __restrict__ h, const float* __restrict__ w2,
                        float* __restrict__ m2, int n) {
    int i = blockIdx.x * blockDim.x + threadIdx.x;
    if (i < n) {
        const float* hr = h + (size_t)i * F_HID;
        float o0 = 0.f, o1 = 0.f, o2 = 0.f;
#pragma unroll
        for (int k = 0; k < F_HID; ++k) {
            float v = hr[k];
            o0 += v * w2[k * F_OUT + 0];
            o1 += v * w2[k * F_OUT + 1];
            o2 += v * w2[k * F_OUT + 2];
        }
        m2[(size_t)i * F_OUT + 0] = o0;
        m2[(size_t)i * F_OUT + 1] = o1;
        m2[(size_t)i * F_OUT + 2] = o2;
    }
}

// ---------------- scatter conv2 (3 features) ----------------
__global__ void k_selfloop3(const float* __restrict__ m2, const float* __restrict__ dinv,
                            float* __restrict__ out, int n) {
    int i = blockIdx.x * blockDim.x + threadIdx.x;
    if (i < n) {
        float w = dinv[i] * dinv[i];
        out[(size_t)i * F_OUT + 0] = m2[(size_t)i * F_OUT + 0] * w;
        out[(size_t)i * F_OUT + 1] = m2[(size_t)i * F_OUT + 1] * w;
        out[(size_t)i * F_OUT + 2] = m2[(size_t)i * F_OUT + 2] * w;
    }
}

__global__ void k_edge3(const long long* __restrict__ src, const long long* __restrict__ dst,
                        const float* __restrict__ m2, const float* __restrict__ dinv,
                        float* __restrict__ out, int e) {
    int idx = blockIdx.x * blockDim.x + threadIdx.x;
    if (idx < e) {
        long long s = src[idx], d = dst[idx];
        float w = dinv[s] * dinv[d];
        float* op = out + (size_t)d * F_OUT;
        const float* mp = m2 + (size_t)s * F_OUT;
        gatomic_add(op + 0, mp[0] * w);
        gatomic_add(op + 1, mp[1] * w);
        gatomic_add(op + 2, mp[2] * w);
    }
}

__global__ void k_logsoftmax(float* __restrict__ out, const float* __restrict__ b2, int n) {
    int i = blockIdx.x * blockDim.x + threadIdx.x;
    if (i < n) {
        float* op = out + (size_t)i * F_OUT;
        float z0 = op[0] + b2[0];
        float z1 = op[1] + b2[1];
        float z2 = op[2] + b2[2];
        float m = fmaxf(z0, fmaxf(z1, z2));
        float l = logf(expf(z0 - m) + expf(z1 - m) + expf(z2 - m));
        op[0] = z0 - m - l;
        op[1] = z1 - m - l;
        op[2] = z2 - m - l;
    }
}

extern "C" void kernel_launch(void* const* d_in, const int* in_sizes, int n_in,
                              void* d_out, int out_size, void* d_ws, size_t ws_size,
                              hipStream_t stream) {
    const float* x      = (const float*)d_in[0];
    const float* W1     = (const float*)d_in[1];
    const float* b1     = (const float*)d_in[2]; (void)b1;  // cancels exactly under BN
    const float* gamma  = (const float*)d_in[3];
    const float* beta   = (const float*)d_in[4];
    const float* W2     = (const float*)d_in[5];
    const float* b2     = (const float*)d_in[6];
    const long long* ei = (const long long*)d_in[7];

    int n = in_sizes[0] / F_IN;        // 200000
    int e = in_sizes[7] / 2;           // 6400000
    const long long* srcp = ei;        // edge_index[0]
    const long long* dstp = ei + e;    // edge_index[1]

    float* ws   = (float*)d_ws;
    float* dinv = ws;                            // n floats (deg -> dinv in place)
    float* m1   = dinv + n;                      // n * 16
    float* agg1 = m1 + (size_t)n * F_HID;        // n * 16 (becomes h after BN+ReLU)
    float* m2   = agg1 + (size_t)n * F_HID;      // n * 3
    float* sums = m2 + (size_t)n * F_OUT;        // 32 floats
    float* outp = (float*)d_out;

    const int B = 256;
    auto cdiv = [](long long a, long long b) { return (int)((a + b - 1) / b); };

    // degree + D^-1/2
    k_init_deg<<<cdiv(n, B), B, 0, stream>>>(dinv, n);
    k_accum_deg<<<cdiv(e, B), B, 0, stream>>>(dstp, dinv, e);
    k_dinv<<<cdiv(n, B), B, 0, stream>>>(dinv, n);

    // conv1: transform (WMMA) + normalized scatter-sum
    int nTiles = cdiv(n, 16);                    // 12500, all full tiles
    k_gemm1_wmma<<<cdiv((long long)nTiles * 32, B), B, 0, stream>>>(x, W1, m1, n, nTiles);
    k_selfloop16<<<cdiv((long long)n * 4, B), B, 0, stream>>>(m1, dinv, agg1, n);
    k_edge16<<<cdiv((long long)e * 4, B), B, 0, stream>>>(srcp, dstp, m1, dinv, agg1, e);

    // batchnorm + relu (in place on agg1)
    k_zero32<<<1, 32, 0, stream>>>(sums);
    k_bn_stats<<<1024, B, 0, stream>>>(agg1, sums, n);
    k_bn_relu<<<cdiv((long long)n * 4, B), B, 0, stream>>>(agg1, sums, gamma, beta, n, 1.0f / (float)n);

    // conv2 + log_softmax (agg2 lives in d_out)
    k_gemm2<<<cdiv(n, B), B, 0, stream>>>(agg1, W2, m2, n);
    k_selfloop3<<<cdiv(n, B), B, 0, stream>>>(m2, dinv, outp, n);
    k_edge3<<<cdiv(e, B), B, 0, stream>>>(srcp, dstp, m2, dinv, outp, e);
    k_logsoftmax<<<cdiv(n, B), B, 0, stream>>>(outp, b2, n);
}